// Net_43207370998397
// MI455X (gfx1250) — compile-verified
//
#include <hip/hip_runtime.h>
#include <math.h>

#define HID 64
#define NEG_SLOPE 0.01f
#define WT_STRIDE 66  // transposed-W LDS row stride: even (keeps b64 alignment) and
                      // coprime-ish with 64 banks (2c mod 64 distinct) -> no conflicts

typedef float v2f __attribute__((ext_vector_type(2)));
typedef float v8f __attribute__((ext_vector_type(8)));

// ---------------------------------------------------------------------------
// Degree / normalization
// ---------------------------------------------------------------------------
__global__ __launch_bounds__(256) void k_deg_init(float* __restrict__ deg, int n) {
    int i = blockIdx.x * 256 + threadIdx.x;
    if (i < n) deg[i] = 1.0f;  // self loop contributes 1
}

__global__ __launch_bounds__(256) void k_deg_count(const int* __restrict__ dst,
                                                   float* __restrict__ deg, int n_edges) {
    int i = blockIdx.x * 256 + threadIdx.x;
    if (i < n_edges) atomicAdd(deg + dst[i], 1.0f);
}

__global__ __launch_bounds__(256) void k_deg_rsqrt(float* __restrict__ deg, int n) {
    int i = blockIdx.x * 256 + threadIdx.x;
    if (i < n) {
        float d = deg[i];
        deg[i] = d > 0.0f ? rsqrtf(d) : 0.0f;
    }
}

// ---------------------------------------------------------------------------
// Layer-1 transform: (N x 10) @ (10 x 64). K=10 not WMMA-shaped; 64M scalar
// FMAs is negligible next to edge traffic.
// ---------------------------------------------------------------------------
__global__ __launch_bounds__(256) void k_gemm_in(const float* __restrict__ X,
                                                 const float* __restrict__ W,
                                                 float* __restrict__ out, int n) {
    int i = blockIdx.x * 256 + threadIdx.x;
    if (i >= n * HID) return;
    int node = i >> 6;
    int f    = i & 63;
    const float* xr = X + node * 10;
    float acc = 0.0f;
#pragma unroll
    for (int k = 0; k < 10; ++k) acc = fmaf(xr[k], W[k * HID + f], acc);
    out[i] = acc;
}

// ---------------------------------------------------------------------------
// Dense 64x64 transform via V_WMMA_F32_16X16X4_F32 (fp32 WMMA, wave32).
// W is staged TRANSPOSED in LDS so each B fragment (two K-adjacent elements of
// one column) is a single aligned ds_load_b64 straight into an even VGPR pair.
// One wave = one 16-row stripe; K-outer loop, 4 column-tile accumulators.
// ---------------------------------------------------------------------------
__global__ __launch_bounds__(256) void k_gemm_wmma(const float* __restrict__ A,
                                                   const float* __restrict__ W,
                                                   float* __restrict__ C, int n) {
    __shared__ float sWt[HID * WT_STRIDE];
    // Transposed fill: thread i handles (col = i>>6, k = i&63); strided global
    // reads of the 16 KB W hit L2, LDS writes are conflict-free (stride 66).
    for (int i = threadIdx.x; i < HID * HID; i += 256) {
        int col = i >> 6;
        int k   = i & 63;
        sWt[col * WT_STRIDE + k] = W[k * HID + col];
    }
    __syncthreads();

    const int wave = threadIdx.x >> 5;
    const int lane = threadIdx.x & 31;
    const int half = lane >> 4;  // 0: lanes 0-15, 1: lanes 16-31
    const int r    = lane & 15;

    const int row0 = blockIdx.x * 128 + wave * 16;
    if (row0 + 16 > n) return;  // wave-uniform exit; active waves keep EXEC all-1s

    // A 16x4 f32 fragment: VGPR v, lane half h -> K = 2*h + v, M = r.
    const float* Arow = A + (size_t)(row0 + r) * HID + 2 * half;
    v2f afrag[16];
#pragma unroll
    for (int j = 0; j < 16; ++j)
        afrag[j] = *(const v2f*)(Arow + 4 * j);

    v8f acc0 = {0.f,0.f,0.f,0.f,0.f,0.f,0.f,0.f};
    v8f acc1 = acc0, acc2 = acc0, acc3 = acc0;

#pragma unroll
    for (int j = 0; j < 16; ++j) {
        const int k = 4 * j + 2 * half;  // this lane's K base for chunk j
        // B 4x16 f32 fragment: VGPR v, lane half h -> K = 2*h + v, N = r.
        // Transposed LDS => both elements contiguous: one b64 load each.
        v2f b0 = *(const v2f*)(sWt + ( 0 + r) * WT_STRIDE + k);
        v2f b1 = *(const v2f*)(sWt + (16 + r) * WT_STRIDE + k);
        v2f b2 = *(const v2f*)(sWt + (32 + r) * WT_STRIDE + k);
        v2f b3 = *(const v2f*)(sWt + (48 + r) * WT_STRIDE + k);
        acc0 = __builtin_amdgcn_wmma_f32_16x16x4_f32(false, afrag[j], false, b0, (short)0, acc0, false, false);
        acc1 = __builtin_amdgcn_wmma_f32_16x16x4_f32(false, afrag[j], false, b1, (short)0, acc1, false, false);
        acc2 = __builtin_amdgcn_wmma_f32_16x16x4_f32(false, afrag[j], false, b2, (short)0, acc2, false, false);
        acc3 = __builtin_amdgcn_wmma_f32_16x16x4_f32(false, afrag[j], false, b3, (short)0, acc3, false, false);
    }

    // C/D 16x16 f32: VGPR v, lane half h -> M = v + 8*h, N = r.
#pragma unroll
    for (int v = 0; v < 8; ++v) {
        float* crow = C + (size_t)(row0 + v + 8 * half) * HID + r;
        crow[ 0] = acc0[v];
        crow[16] = acc1[v];
        crow[32] = acc2[v];
        crow[48] = acc3[v];
    }
}

// ---------------------------------------------------------------------------
// Aggregation. h (25.6 MB) is L2-resident (192 MB L2): gather + f32 atomics
// resolve in L2. Atomic count (E*64/layer) is the floor (no packed f32 global
// atomic exists); gather side vectorized to float4.
// ---------------------------------------------------------------------------
__global__ __launch_bounds__(256) void k_self_init4(const float4* __restrict__ tmp,
                                                    const float* __restrict__ dinv,
                                                    float4* __restrict__ out, int n) {
    int i = blockIdx.x * 256 + threadIdx.x;
    if (i >= n * (HID / 4)) return;
    float  di = dinv[i >> 4];
    float  s  = di * di;
    float4 t  = tmp[i];
    out[i] = make_float4(t.x * s, t.y * s, t.z * s, t.w * s);
}

__global__ __launch_bounds__(256) void k_edge_agg(const float4* __restrict__ tmp,
                                                  const int* __restrict__ src,
                                                  const int* __restrict__ dst,
                                                  const float* __restrict__ dinv,
                                                  float* __restrict__ out, int n_edges) {
    int idx = blockIdx.x * 256 + threadIdx.x;
    int e   = idx >> 4;   // 16 threads per edge
    int f4  = idx & 15;   // float4 chunk within the 64-feature row
    if (e >= n_edges) return;
    int    s    = src[e];
    int    d    = dst[e];
    float  norm = dinv[s] * dinv[d];
    float4 v    = tmp[(size_t)s * (HID / 4) + f4];  // global_load_b128, coalesced
    float* o    = out + (size_t)d * HID + f4 * 4;
    atomicAdd(o + 0, v.x * norm);
    atomicAdd(o + 1, v.y * norm);
    atomicAdd(o + 2, v.z * norm);
    atomicAdd(o + 3, v.w * norm);
}

__global__ __launch_bounds__(256) void k_bias_lrelu4(float4* __restrict__ h,
                                                     const float* __restrict__ bias, int n) {
    int i = blockIdx.x * 256 + threadIdx.x;
    if (i >= n * (HID / 4)) return;
    float4 b = ((const float4*)bias)[i & 15];
    float4 v = h[i];
    v.x += b.x; v.y += b.y; v.z += b.z; v.w += b.w;
    v.x = v.x > 0.f ? v.x : NEG_SLOPE * v.x;
    v.y = v.y > 0.f ? v.y : NEG_SLOPE * v.y;
    v.z = v.z > 0.f ? v.z : NEG_SLOPE * v.z;
    v.w = v.w > 0.f ? v.w : NEG_SLOPE * v.w;
    h[i] = v;
}

// ---------------------------------------------------------------------------
// Final FC: (N x 64) @ (64 x 10) + bias. 64M MACs, scalar.
// ---------------------------------------------------------------------------
__global__ __launch_bounds__(256) void k_fc(const float* __restrict__ H,
                                            const float* __restrict__ W,
                                            const float* __restrict__ b,
                                            float* __restrict__ out, int n) {
    int i = blockIdx.x * 256 + threadIdx.x;
    if (i >= n * 10) return;
    int node = i / 10;
    int c    = i % 10;
    const float* hr = H + (size_t)node * HID;
    float acc = b[c];
#pragma unroll
    for (int k = 0; k < HID; ++k) acc = fmaf(hr[k], W[k * 10 + c], acc);
    out[i] = acc;
}

// ---------------------------------------------------------------------------
extern "C" void kernel_launch(void* const* d_in, const int* in_sizes, int n_in,
                              void* d_out, int out_size, void* d_ws, size_t ws_size,
                              hipStream_t stream) {
    const float* x   = (const float*)d_in[0];
    const int*   ei  = (const int*)d_in[1];
    const float* W1  = (const float*)d_in[2];
    const float* b1  = (const float*)d_in[3];
    const float* W2  = (const float*)d_in[4];
    const float* b2  = (const float*)d_in[5];
    const float* W3  = (const float*)d_in[6];
    const float* b3  = (const float*)d_in[7];
    const float* Wfc = (const float*)d_in[8];
    const float* bfc = (const float*)d_in[9];

    const int n_nodes = in_sizes[0] / 10;
    const int n_edges = in_sizes[1] / 2;
    const int* src = ei;
    const int* dst = ei + n_edges;

    // Workspace carve: dinv | bufA (N x 64) | bufB (N x 64)  (~52 MB)
    float* dinv = (float*)d_ws;
    float* bufA = dinv + (((size_t)n_nodes + 255) & ~(size_t)255);
    float* bufB = bufA + (size_t)n_nodes * HID;

    const int gN  = (n_nodes + 255) / 256;                 // node-parallel
    const int gE  = (n_edges + 255) / 256;                 // edge-parallel
    const int gNF = (n_nodes * HID + 255) / 256;           // node*feature (scalar)
    const int gN4 = (n_nodes * (HID / 4) + 255) / 256;     // node*feature (float4)
    const int gEA = (n_edges * 16 + 255) / 256;            // 16 threads / edge
    const int gW  = ((n_nodes + 15) / 16 + 7) / 8;         // 16-row tiles, 8 waves/block
    const int gFC = (n_nodes * 10 + 255) / 256;

    // --- symmetric normalization: dinv = rsqrt(1 + indegree) ---
    k_deg_init <<<gN, 256, 0, stream>>>(dinv, n_nodes);
    k_deg_count<<<gE, 256, 0, stream>>>(dst, dinv, n_edges);
    k_deg_rsqrt<<<gN, 256, 0, stream>>>(dinv, n_nodes);

    // --- layer 1: transform (scalar, K=10) -> aggregate -> bias+leakyrelu ---
    k_gemm_in    <<<gNF, 256, 0, stream>>>(x, W1, bufB, n_nodes);
    k_self_init4 <<<gN4, 256, 0, stream>>>((const float4*)bufB, dinv, (float4*)bufA, n_nodes);
    k_edge_agg   <<<gEA, 256, 0, stream>>>((const float4*)bufB, src, dst, dinv, bufA, n_edges);
    k_bias_lrelu4<<<gN4, 256, 0, stream>>>((float4*)bufA, b1, n_nodes);

    // --- layer 2: WMMA transform -> aggregate -> bias+leakyrelu ---
    k_gemm_wmma  <<<gW, 256, 0, stream>>>(bufA, W2, bufB, n_nodes);
    k_self_init4 <<<gN4, 256, 0, stream>>>((const float4*)bufB, dinv, (float4*)bufA, n_nodes);
    k_edge_agg   <<<gEA, 256, 0, stream>>>((const float4*)bufB, src, dst, dinv, bufA, n_edges);
    k_bias_lrelu4<<<gN4, 256, 0, stream>>>((float4*)bufA, b2, n_nodes);

    // --- layer 3 ---
    k_gemm_wmma  <<<gW, 256, 0, stream>>>(bufA, W3, bufB, n_nodes);
    k_self_init4 <<<gN4, 256, 0, stream>>>((const float4*)bufB, dinv, (float4*)bufA, n_nodes);
    k_edge_agg   <<<gEA, 256, 0, stream>>>((const float4*)bufB, src, dst, dinv, bufA, n_edges);
    k_bias_lrelu4<<<gN4, 256, 0, stream>>>((float4*)bufA, b3, n_nodes);

    // --- classifier head ---
    k_fc<<<gFC, 256, 0, stream>>>(bufA, Wfc, bfc, (float*)d_out, n_nodes);
}